// GCN3Classifier_35021163332022
// MI455X (gfx1250) — compile-verified
//
#include <hip/hip_runtime.h>
#include <hip/hip_bf16.h>

typedef __attribute__((ext_vector_type(2))) float v2f;
typedef __attribute__((ext_vector_type(8))) float v8f;

#define NNODES 100000
#define NEDGES 1600000

// ---------------------------------------------------------------------------
// CSR construction: counts (with self-loop), scan -> row_ptr + dinv,
// atomic scatter fill, per-row canonical sort (determinism).
// ---------------------------------------------------------------------------

__global__ void init_counts_kernel(int* __restrict__ cnt, int* __restrict__ cursor, int n) {
    int i = blockIdx.x * blockDim.x + threadIdx.x;
    if (i < n) { cnt[i] = 1; cursor[i] = 0; }   // 1 = self-loop
}

__global__ void count_edges_kernel(const int* __restrict__ ei, int e, int* __restrict__ cnt) {
    int t = blockIdx.x * blockDim.x + threadIdx.x;
    if (t < e) atomicAdd(&cnt[ei[e + t]], 1);   // dst row of edge_index (2,E)
}

// Single-workgroup chunked Hillis-Steele exclusive scan; also emits dinv.
__global__ void scan_kernel(const int* __restrict__ cnt, int* __restrict__ rowptr,
                            float* __restrict__ dinv, int n) {
    __shared__ int sdata[1024];
    __shared__ int s_carry;
    const int t = threadIdx.x;
    if (t == 0) s_carry = 0;
    __syncthreads();
    for (int base = 0; base < n; base += 1024) {
        int idx = base + t;
        int v = (idx < n) ? cnt[idx] : 0;
        sdata[t] = v;
        __syncthreads();
        for (int off = 1; off < 1024; off <<= 1) {
            int x = (t >= off) ? sdata[t - off] : 0;
            __syncthreads();
            sdata[t] += x;
            __syncthreads();
        }
        int incl = sdata[t];
        int carry = s_carry;
        if (idx < n) {
            rowptr[idx] = carry + incl - v;         // exclusive
            dinv[idx]   = rsqrtf((float)v);         // deg >= 1 (self-loop)
        }
        __syncthreads();
        if (t == 1023) s_carry = carry + sdata[1023];
        __syncthreads();
    }
    if (t == 0) rowptr[n] = s_carry;
}

__global__ void fill_csr_kernel(const int* __restrict__ ei, int e, int n,
                                const int* __restrict__ rowptr,
                                int* __restrict__ cursor, int* __restrict__ col) {
    int t = blockIdx.x * blockDim.x + threadIdx.x;
    if (t < e) {
        int s = ei[t];
        int d = ei[e + t];
        int pos = rowptr[d] + atomicAdd(&cursor[d], 1);
        col[pos] = s;
    } else if (t < e + n) {
        int i = t - e;                               // self-loop entry
        int pos = rowptr[i] + atomicAdd(&cursor[i], 1);
        col[pos] = i;
    }
}

// Canonicalize neighbor order within each row (avg degree ~17) so the
// float summation order is deterministic across runs despite atomic fill.
__global__ void sort_rows_kernel(const int* __restrict__ rowptr, int* __restrict__ col, int n) {
    int i = blockIdx.x * blockDim.x + threadIdx.x;
    if (i >= n) return;
    int s = rowptr[i], e = rowptr[i + 1];
    for (int a = s + 1; a < e; ++a) {
        int key = col[a];
        int b = a - 1;
        while (b >= s && col[b] > key) { col[b + 1] = col[b]; --b; }
        col[b + 1] = key;
    }
}

// ---------------------------------------------------------------------------
// Dense transform H = X * W using V_WMMA_F32_16X16X4_F32 (exact f32 path).
// One wave owns a 16-row tile and all output columns.
//
// W is staged in LDS in *fragment-major* layout: element (k, c) lives at
//   sW[(k/2)*S + c*2 + (k&1)]
// so a lane's B fragment (B[k][c], B[k+1][c]) is a single aligned b64 LDS
// load directly into an even VGPR pair (no repack movs). Row stride S is
// padded so consecutive k-pair rows differ by 128 B (mod 256 B): the two
// half-waves (lanes 0-15 row k0/2, lanes 16-31 row k0/2+1) then cover
// disjoint halves of the 64 x 4 B bank window -> conflict-free.
//
// Grid is sized so every wave owns a full valid tile (EXEC all-1s at every
// WMMA, per ISA restriction).
//
// Register layouts per ISA 7.12.2 (32-bit):
//   A 16x4 : lanes 0-15 -> VGPR0=K0,VGPR1=K1 ; lanes 16-31 -> K2,K3
//   B 4x16 : lanes 0-15 -> VGPR0=B[K0][n],VGPR1=B[K1][n]; lanes16-31 -> K2,K3
//   C 16x16: VGPR r -> rows r (lanes 0-15) and r+8 (lanes 16-31), N = lane%16
// ---------------------------------------------------------------------------

template <int INF, int OUTF>
struct GemmCfg {
    static constexpr int OUTW = (OUTF < 16) ? 16 : OUTF;   // padded cols
    static constexpr int NCT  = OUTW / 16;                 // col tiles
    static constexpr int S0   = OUTW * 2;                  // floats per k-pair row
    static constexpr int S    = (S0 % 64 == 32) ? S0 : S0 + 32;  // S*4 % 256 == 128
    static constexpr size_t ldsBytes = (size_t)(INF / 2) * S * sizeof(float);
};

template <int INF, int OUTF>
__global__ void gemm_wmma_kernel(const float* __restrict__ X, const float* __restrict__ W,
                                 float* __restrict__ H, int nRowTiles) {
    using C = GemmCfg<INF, OUTF>;
    extern __shared__ float sW[];   // (INF/2) x C::S floats, fragment-major

    // Cooperative swizzled fill (zero-pad columns OUTF..OUTW).
    for (int idx = threadIdx.x; idx < INF * C::OUTW; idx += blockDim.x) {
        int k = idx / C::OUTW;
        int c = idx % C::OUTW;
        float v = (c < OUTF) ? W[k * OUTF + c] : 0.0f;
        sW[(k >> 1) * C::S + c * 2 + (k & 1)] = v;
    }
    __syncthreads();

    const int wave = threadIdx.x >> 5;
    const int lane = threadIdx.x & 31;
    const int tile = blockIdx.x * (blockDim.x >> 5) + wave;   // always < nRowTiles
    (void)nRowTiles;

    const int half = lane >> 4;       // 0: K pair (k0,k0+1), 1: K pair (k0+2,k0+3)
    const int m    = lane & 15;       // row (for A) / column (for B, C)
    const float* __restrict__ xrow = X + (size_t)(tile * 16 + m) * INF;

    v8f acc[C::NCT];
#pragma unroll
    for (int ct = 0; ct < C::NCT; ++ct)
#pragma unroll
        for (int q = 0; q < 8; ++q) acc[ct][q] = 0.0f;

    for (int k0 = 0; k0 < INF; k0 += 4) {
        // A fragment: one aligned b64 global load per lane.
        v2f a = *reinterpret_cast<const v2f*>(xrow + k0 + 2 * half);
        const float* brow = sW + (k0 / 2 + half) * C::S;
#pragma unroll
        for (int ct = 0; ct < C::NCT; ++ct) {
            // B fragment: one aligned b64 LDS load per lane.
            v2f b = *reinterpret_cast<const v2f*>(brow + (ct * 16 + m) * 2);
            acc[ct] = __builtin_amdgcn_wmma_f32_16x16x4_f32(
                false, a, false, b, (short)0, acc[ct], false, false);
        }
    }

#pragma unroll
    for (int ct = 0; ct < C::NCT; ++ct) {
        const int c = ct * 16 + m;
#pragma unroll
        for (int r = 0; r < 8; ++r) {
            int row = tile * 16 + r + 8 * half;
            if (OUTF >= 16 || c < OUTF)
                H[(size_t)row * OUTF + c] = acc[ct][r];
        }
    }
}

// ---------------------------------------------------------------------------
// Aggregation: out[i,f] = act( dinv[i] * sum_{j in row i} dinv[j]*h[j,f] + b[f] )
// One thread per (node, feature): f contiguous -> coalesced h loads; col/dinv
// broadcast within the F-thread group and L2-resident (h1/h2 fit in 192MB L2).
// ---------------------------------------------------------------------------

template <int F, bool RELU>
__global__ void agg_kernel(const float* __restrict__ h, const int* __restrict__ rowptr,
                           const int* __restrict__ col, const float* __restrict__ dinv,
                           const float* __restrict__ bias, float* __restrict__ out, int n) {
    int gid = blockIdx.x * blockDim.x + threadIdx.x;
    int i = gid / F;
    int f = gid % F;
    if (i >= n) return;
    int s = rowptr[i], e = rowptr[i + 1];
    float acc = 0.0f;
    for (int p = s; p < e; ++p) {
        int j = col[p];
        acc = fmaf(dinv[j], h[(size_t)j * F + f], acc);
    }
    float v = fmaf(dinv[i], acc, bias[f]);
    if (RELU) v = fmaxf(v, 0.0f);
    out[(size_t)i * F + f] = v;
}

// ---------------------------------------------------------------------------

static inline size_t align_up(size_t v, size_t a) { return (v + a - 1) & ~(a - 1); }

extern "C" void kernel_launch(void* const* d_in, const int* in_sizes, int n_in,
                              void* d_out, int out_size, void* d_ws, size_t ws_size,
                              hipStream_t stream) {
    const float* x   = (const float*)d_in[0];
    const int*   ei  = (const int*)d_in[1];     // (2,E): [0,E)=src, [E,2E)=dst
    const float* W1  = (const float*)d_in[2];
    const float* b1  = (const float*)d_in[3];
    const float* W2  = (const float*)d_in[4];
    const float* b2  = (const float*)d_in[5];
    const float* W3  = (const float*)d_in[6];
    const float* b3  = (const float*)d_in[7];
    float* out = (float*)d_out;

    const int n = NNODES;
    const int e = NEDGES;
    const int nnz = e + n;

    // Workspace bump allocation
    char* ws = (char*)d_ws;
    size_t off = 0;
    int*   cnt    = (int*)(ws + off);  off = align_up(off + (size_t)n * 4, 256);
    int*   cursor = (int*)(ws + off);  off = align_up(off + (size_t)n * 4, 256);
    int*   rowptr = (int*)(ws + off);  off = align_up(off + (size_t)(n + 1) * 4, 256);
    float* dinv   = (float*)(ws + off); off = align_up(off + (size_t)n * 4, 256);
    int*   col    = (int*)(ws + off);  off = align_up(off + (size_t)nnz * 4, 256);
    float* bufA   = (float*)(ws + off); off = align_up(off + (size_t)n * 64 * 4, 256);
    float* bufB   = (float*)(ws + off); off = align_up(off + (size_t)n * 64 * 4, 256);
    (void)ws_size; (void)n_in; (void)in_sizes; (void)out_size;

    const int T = 256;
    // --- CSR build (shared across all three layers, like cached gcn_norm) ---
    init_counts_kernel<<<(n + T - 1) / T, T, 0, stream>>>(cnt, cursor, n);
    count_edges_kernel<<<(e + T - 1) / T, T, 0, stream>>>(ei, e, cnt);
    scan_kernel<<<1, 1024, 0, stream>>>(cnt, rowptr, dinv, n);
    fill_csr_kernel<<<(nnz + T - 1) / T, T, 0, stream>>>(ei, e, n, rowptr, cursor, col);
    sort_rows_kernel<<<(n + T - 1) / T, T, 0, stream>>>(rowptr, col, n);

    // 100000 rows = 6250 row-tiles of 16; 10 waves/block -> 625 full blocks.
    const int rowTiles = n / 16;            // 6250
    const int gemmBlk  = 320;               // 10 waves
    const int gemmGrid = rowTiles / 10;     // 625, exact

    // --- Layer 1: 512 -> 64 ---
    gemm_wmma_kernel<512, 64><<<gemmGrid, gemmBlk, GemmCfg<512, 64>::ldsBytes, stream>>>(
        x, W1, bufA, rowTiles);
    agg_kernel<64, true><<<((size_t)n * 64 + T - 1) / T, T, 0, stream>>>(
        bufA, rowptr, col, dinv, b1, bufB, n);

    // --- Layer 2: 64 -> 32 ---
    gemm_wmma_kernel<64, 32><<<gemmGrid, gemmBlk, GemmCfg<64, 32>::ldsBytes, stream>>>(
        bufB, W2, bufA, rowTiles);
    agg_kernel<32, true><<<((size_t)n * 32 + T - 1) / T, T, 0, stream>>>(
        bufA, rowptr, col, dinv, b2, bufB, n);

    // --- Layer 3: 32 -> 4 (B zero-padded to 16 cols in LDS) ---
    gemm_wmma_kernel<32, 4><<<gemmGrid, gemmBlk, GemmCfg<32, 4>::ldsBytes, stream>>>(
        bufB, W3, bufA, rowTiles);
    agg_kernel<4, false><<<((size_t)n * 4 + T - 1) / T, T, 0, stream>>>(
        bufA, rowptr, col, dinv, b3, out, n);
}